// I2SBDiffusion_34153579938495
// MI455X (gfx1250) — compile-verified
//
#include <hip/hip_runtime.h>
#include <math.h>

#define GG 8192   // graphs per batch (fixed by the reference)

typedef float v2f __attribute__((ext_vector_type(2)));
typedef float v8f __attribute__((ext_vector_type(8)));

// ---------------------------------------------------------------------------
// K0: diffusion schedule from beta[T]  (serial: T=1000, negligible)
// sched layout: [0,T)=mu_x0  [T,2T)=mu_x1  [2T,3T)=std_sb  [3T,4T)=std_fwd
// ---------------------------------------------------------------------------
__global__ void i2sb_sched(const float* __restrict__ beta,
                           float* __restrict__ sched, int T)
{
  if (threadIdx.x != 0 || blockIdx.x != 0) return;
  float total = 0.f;
  for (int t = 0; t < T; ++t) total += beta[t];
  float* c0 = sched;          // mu_x0 = std_fwd^2/denom
  float* c1 = sched + T;      // mu_x1 = std_bwd^2/denom
  float* cs = sched + 2 * T;  // std_sb
  float* sf = sched + 3 * T;  // std_fwd
  float pf = 0.f;
  for (int t = 0; t < T; ++t) {
    float b = beta[t];
    pf += b;                       // inclusive prefix = std_fwd^2
    float sf2 = pf;
    float sb2 = total - pf + b;    // reversed-cumsum = std_bwd^2
    float den = sf2 + sb2;
    c0[t] = sf2 / den;
    c1[t] = sb2 / den;
    cs[t] = sqrtf(sf2 * sb2 / den);
    sf[t] = sqrtf(sf2);
  }
}

// ---------------------------------------------------------------------------
// K1: x_t = mu0*x0 + mu1*x1 + ssb*noise   (4 nodes / thread, 128-bit I/O)
// ---------------------------------------------------------------------------
__global__ void i2sb_xt(const float4* __restrict__ x0,
                        const float4* __restrict__ x1,
                        const float4* __restrict__ nz,
                        const int* __restrict__ batch,
                        const int* __restrict__ timev,
                        const float* __restrict__ sched,
                        float4* __restrict__ xt, int N, int T)
{
  int i4 = blockIdx.x * blockDim.x + threadIdx.x;
  int nq = N >> 2;
  if (i4 >= nq) return;
  const float* c0 = sched;
  const float* c1 = sched + T;
  const float* cs = sched + 2 * T;
  __attribute__((aligned(16))) float fa[12], fb[12], fn[12], fo[12];
  {
    float4* pa = (float4*)fa; float4* pb = (float4*)fb; float4* pn = (float4*)fn;
#pragma unroll
    for (int q = 0; q < 3; ++q) {
      pa[q] = x0[i4 * 3 + q];
      pb[q] = x1[i4 * 3 + q];
      pn[q] = nz[i4 * 3 + q];
    }
  }
#pragma unroll
  for (int j = 0; j < 4; ++j) {
    int node = i4 * 4 + j;
    int g = batch[node];
    int t = timev[g];                 // idx = time[batch[i]]
    float m0 = c0[t], m1 = c1[t], s = cs[t];
#pragma unroll
    for (int k = 0; k < 3; ++k)
      fo[3 * j + k] = m0 * fa[3 * j + k] + m1 * fb[3 * j + k] + s * fn[3 * j + k];
  }
  {
    const float4* po = (const float4*)fo;
#pragma unroll
    for (int q = 0; q < 3; ++q) xt[i4 * 3 + q] = po[q];
  }
}

// ---------------------------------------------------------------------------
// K2: per-graph segment starts from the *sorted* batch vector
// start[g] = first node of graph g; start[GG] = N
// ---------------------------------------------------------------------------
__global__ void i2sb_offs(const int* __restrict__ batch,
                          int* __restrict__ start, int N)
{
  int i = blockIdx.x * blockDim.x + threadIdx.x;
  if (i >= N) return;
  int gi = batch[i];
  int gp = (i == 0) ? -1 : batch[i - 1];
  for (int g = gp + 1; g <= gi; ++g) start[g] = i;
  if (i == N - 1)
    for (int g = gi + 1; g <= GG; ++g) start[g] = N;
}

__device__ __forceinline__ float det3f(const float M[3][3]) {
  return M[0][0] * (M[1][1] * M[2][2] - M[1][2] * M[2][1])
       - M[0][1] * (M[1][0] * M[2][2] - M[1][2] * M[2][0])
       + M[0][2] * (M[1][0] * M[2][1] - M[1][1] * M[2][0]);
}

// ---------------------------------------------------------------------------
// K3: one block (8 wave32) per graph.
//  M4 = sum_i (x0_i,1) (xt_i,1)^T via V_WMMA_F32_16X16X4_F32 (4 nodes/issue)
//  -> cnt, COMs and H = S3 - sx0*sxt^T/cnt in one pass, no atomics.
//  All lane masking is done with value selects, never EXEC branches, so the
//  wave keeps EXEC==all-ones around every WMMA (ISA requirement) and the
//  inner loop stays free of saveexec round-trips.
//  Thread 0 then runs a 3x3 Jacobi-SVD Kabsch and stores R, com_x0, com_xt.
// ---------------------------------------------------------------------------
__global__ __launch_bounds__(256) void i2sb_kabsch(
    const float* __restrict__ x0, const float* __restrict__ xt,
    const int* __restrict__ start,
    float* __restrict__ Rm, float* __restrict__ cA, float* __restrict__ cB)
{
  __shared__ float stage[8][32][8];   // per-wave: 32 nodes x (a[4] | b[4])
  __shared__ float red[8][16];        // per-wave 4x4 partial (col-major)
  int g = blockIdx.x;
  int s = start[g];
  int e = start[g + 1];
  int n = e - s;
  int wave = threadIdx.x >> 5;
  int lane = threadIdx.x & 31;
  int m   = lane & 15;
  bool mok = (m < 4);                 // only M/N rows 0..3 are real data
  int kA  = (lane < 16) ? 0 : 2;      // f32 A frag: v0 holds K0/K2, v1 holds K1/K3
  int kB0 = (lane < 16) ? 0 : 1;      // f32 B frag: v0 rows K0/K1, v1 rows K2/K3
  int kB1 = (lane < 16) ? 2 : 3;
  v8f c0v = {0.f, 0.f, 0.f, 0.f, 0.f, 0.f, 0.f, 0.f};
  v8f c1v = {0.f, 0.f, 0.f, 0.f, 0.f, 0.f, 0.f, 0.f};
  int nch = (n + 31) >> 5;            // 32-node chunks, wave-strided (uniform per wave)
  for (int ch = wave; ch < nch; ch += 8) {
    int idx = s + ch * 32 + lane;
    // branch-free tail handling: clamp the address (e-1 >= s >= 0 whenever the
    // loop body runs), load unconditionally, select staged values to zero.
    bool ok = (idx < e);
    int idxc = ok ? idx : (e - 1);
    float ax = x0[3 * idxc + 0], ay = x0[3 * idxc + 1], az = x0[3 * idxc + 2];
    float bx = xt[3 * idxc + 0], by = xt[3 * idxc + 1], bz = xt[3 * idxc + 2];
    float w = ok ? 1.f : 0.f;
    ax = ok ? ax : 0.f; ay = ok ? ay : 0.f; az = ok ? az : 0.f;
    bx = ok ? bx : 0.f; by = ok ? by : 0.f; bz = ok ? bz : 0.f;
    float* st = &stage[wave][lane][0];
    st[0] = ax; st[1] = ay; st[2] = az; st[3] = w;
    st[4] = bx; st[5] = by; st[6] = bz; st[7] = w;
    asm volatile("" ::: "memory");    // same-wave DS is in-order; pin compiler order
#pragma unroll
    for (int j = 0; j < 8; ++j) {     // 8 WMMA = 32 nodes, 2 accumulators
      const float* base = &stage[wave][j * 4][0];
      // Unconditional ds_load (address always inside this block's LDS
      // allocation) + value select; garbage lanes (m>=4) are zeroed.
      float av0 = base[kA * 8 + m];
      float av1 = base[(kA + 1) * 8 + m];
      float bv0 = base[kB0 * 8 + 4 + m];
      float bv1 = base[kB1 * 8 + 4 + m];
      av0 = mok ? av0 : 0.f;
      av1 = mok ? av1 : 0.f;
      bv0 = mok ? bv0 : 0.f;
      bv1 = mok ? bv1 : 0.f;
      v2f af = {av0, av1};
      v2f bf = {bv0, bv1};
      if (j & 1)
        c1v = __builtin_amdgcn_wmma_f32_16x16x4_f32(false, af, false, bf,
                                                    (short)0, c1v, false, false);
      else
        c0v = __builtin_amdgcn_wmma_f32_16x16x4_f32(false, af, false, bf,
                                                    (short)0, c0v, false, false);
    }
    asm volatile("" ::: "memory");
  }
  // C layout: VGPR v, lanes 0-15 -> (M=v, N=lane). We need M,N in 0..3.
  if (lane < 4) {
#pragma unroll
    for (int r = 0; r < 4; ++r)
      red[wave][lane * 4 + r] = c0v[r] + c1v[r];  // col-major M4
  }
  __syncthreads();
  if (threadIdx.x == 0) {
    float M4[16];
    for (int t = 0; t < 16; ++t) {
      float acc = 0.f;
      for (int w = 0; w < 8; ++w) acc += red[w][t];
      M4[t] = acc;
    }
    float cntRaw = M4[15];
    float cnt = fmaxf(cntRaw, 1.f);
    float sx0[3], sxt[3];
    for (int r = 0; r < 3; ++r) sx0[r] = M4[12 + r];       // col 3 = sum a
    for (int q = 0; q < 3; ++q) sxt[q] = M4[q * 4 + 3];    // row 3 = sum b
    float H[3][3];
    float hf = 0.f;
    for (int r = 0; r < 3; ++r)
      for (int q = 0; q < 3; ++q) {
        H[r][q] = M4[q * 4 + r] - sx0[r] * sxt[q] / cnt;   // centered covariance
        hf += H[r][q] * H[r][q];
      }
    float Rr[3][3] = {{1.f,0.f,0.f},{0.f,1.f,0.f},{0.f,0.f,1.f}};
    if (cntRaw > 0.5f && hf > 1e-24f) {
      // --- Jacobi eigensolve of K = H^T H -> V, lambda ---
      float Kx[3][3];
      for (int i = 0; i < 3; ++i)
        for (int j = 0; j < 3; ++j)
          Kx[i][j] = H[0][i]*H[0][j] + H[1][i]*H[1][j] + H[2][i]*H[2][j];
      float V[3][3] = {{1.f,0.f,0.f},{0.f,1.f,0.f},{0.f,0.f,1.f}};
      const int JP[3] = {0, 0, 1};
      const int JQ[3] = {1, 2, 2};
#pragma unroll 1
      for (int sweep = 0; sweep < 10; ++sweep) {
#pragma unroll 1
        for (int pq = 0; pq < 3; ++pq) {
          int p = JP[pq], q = JQ[pq];
          float apq = Kx[p][q];
          if (fabsf(apq) < 1e-28f) continue;
          float theta = 0.5f * (Kx[q][q] - Kx[p][p]) / apq;
          float tt = copysignf(1.f, theta) / (fabsf(theta) + sqrtf(theta * theta + 1.f));
          float cth = 1.f / sqrtf(tt * tt + 1.f);
          float sth = tt * cth;
          float app = Kx[p][p], aqq = Kx[q][q];
          Kx[p][p] = app - tt * apq;
          Kx[q][q] = aqq + tt * apq;
          Kx[p][q] = Kx[q][p] = 0.f;
          int r3 = 3 - p - q;
          float arp = Kx[r3][p], arq = Kx[r3][q];
          Kx[r3][p] = Kx[p][r3] = cth * arp - sth * arq;
          Kx[r3][q] = Kx[q][r3] = sth * arp + cth * arq;
          for (int i = 0; i < 3; ++i) {
            float vip = V[i][p], viq = V[i][q];
            V[i][p] = cth * vip - sth * viq;
            V[i][q] = sth * vip + cth * viq;
          }
        }
      }
      float lam[3] = {Kx[0][0], Kx[1][1], Kx[2][2]};
      int o0 = 0, o1 = 1, o2 = 2, tmpi;
      if (lam[o0] < lam[o1]) { tmpi = o0; o0 = o1; o1 = tmpi; }
      if (lam[o0] < lam[o2]) { tmpi = o0; o0 = o2; o2 = tmpi; }
      if (lam[o1] < lam[o2]) { tmpi = o1; o1 = o2; o2 = tmpi; }
      int ord[3] = {o0, o1, o2};
      float Vs[3][3];
      for (int k = 0; k < 3; ++k)
        for (int i = 0; i < 3; ++i) Vs[i][k] = V[i][ord[k]];
      // --- U columns: u_k = normalize(H v_k), with degenerate fallbacks ---
      float Uc[3][3];
#pragma unroll 1
      for (int k = 0; k < 3; ++k) {
        float u0 = H[0][0]*Vs[0][k] + H[0][1]*Vs[1][k] + H[0][2]*Vs[2][k];
        float u1 = H[1][0]*Vs[0][k] + H[1][1]*Vs[1][k] + H[1][2]*Vs[2][k];
        float u2 = H[2][0]*Vs[0][k] + H[2][1]*Vs[1][k] + H[2][2]*Vs[2][k];
        float len = sqrtf(u0*u0 + u1*u1 + u2*u2);
        if (len > 1e-20f) {
          float il = 1.f / len;
          Uc[0][k] = u0 * il; Uc[1][k] = u1 * il; Uc[2][k] = u2 * il;
        } else if (k == 0) {
          Uc[0][0] = 1.f; Uc[1][0] = 0.f; Uc[2][0] = 0.f;
        } else if (k == 1) {
          float e0 = (fabsf(Uc[0][0]) < 0.9f) ? 1.f : 0.f;
          float e1 = 1.f - e0;
          float w0 = -Uc[2][0] * e1;
          float w1 =  Uc[2][0] * e0;
          float w2 =  Uc[0][0] * e1 - Uc[1][0] * e0;
          float wl = sqrtf(w0*w0 + w1*w1 + w2*w2) + 1e-30f;
          Uc[0][1] = w0/wl; Uc[1][1] = w1/wl; Uc[2][1] = w2/wl;
        } else {
          Uc[0][2] = Uc[1][0]*Uc[2][1] - Uc[2][0]*Uc[1][1];
          Uc[1][2] = Uc[2][0]*Uc[0][1] - Uc[0][0]*Uc[2][1];
          Uc[2][2] = Uc[0][0]*Uc[1][1] - Uc[1][0]*Uc[0][1];
        }
      }
      float dsgn = (det3f(Uc) * det3f(Vs) < 0.f) ? -1.f : 1.f;
      // R = V diag(1,1,d) U^T
      for (int i = 0; i < 3; ++i)
        for (int j = 0; j < 3; ++j)
          Rr[i][j] = Vs[i][0]*Uc[j][0] + Vs[i][1]*Uc[j][1] + dsgn*Vs[i][2]*Uc[j][2];
    }
    float* Rg = Rm + 9 * g;
    for (int i = 0; i < 3; ++i)
      for (int j = 0; j < 3; ++j) Rg[i * 3 + j] = Rr[i][j];
    for (int r = 0; r < 3; ++r) {
      cA[3 * g + r] = sx0[r] / cnt;
      cB[3 * g + r] = sxt[r] / cnt;
    }
  }
}

// ---------------------------------------------------------------------------
// K4: target = (x_t - (R (x0 - com_x0) + com_xt)) / std_fwd[t]
// ---------------------------------------------------------------------------
__global__ void i2sb_target(const float4* __restrict__ x0,
                            const float4* __restrict__ xt,
                            const int* __restrict__ batch,
                            const int* __restrict__ timev,
                            const float* __restrict__ sfwd,
                            const float* __restrict__ Rm,
                            const float* __restrict__ cA,
                            const float* __restrict__ cB,
                            float4* __restrict__ outT, int N)
{
  int i4 = blockIdx.x * blockDim.x + threadIdx.x;
  int nq = N >> 2;
  if (i4 >= nq) return;
  __attribute__((aligned(16))) float fa[12], fb[12], fo[12];
  {
    float4* pa = (float4*)fa; float4* pb = (float4*)fb;
#pragma unroll
    for (int q = 0; q < 3; ++q) { pa[q] = x0[i4 * 3 + q]; pb[q] = xt[i4 * 3 + q]; }
  }
#pragma unroll
  for (int j = 0; j < 4; ++j) {
    int node = i4 * 4 + j;
    int g = batch[node];
    int t = timev[g];
    const float* R = Rm + 9 * g;
    float px = fa[3*j+0] - cA[3*g+0];
    float py = fa[3*j+1] - cA[3*g+1];
    float pz = fa[3*j+2] - cA[3*g+2];
    float inv = 1.f / sfwd[t];
#pragma unroll
    for (int r = 0; r < 3; ++r) {
      float al = R[r*3+0]*px + R[r*3+1]*py + R[r*3+2]*pz + cB[3*g+r];
      fo[3*j+r] = (fb[3*j+r] - al) * inv;
    }
  }
  {
    const float4* po = (const float4*)fo;
#pragma unroll
    for (int q = 0; q < 3; ++q) outT[i4 * 3 + q] = po[q];
  }
}

// ---------------------------------------------------------------------------
extern "C" void kernel_launch(void* const* d_in, const int* in_sizes, int n_in,
                              void* d_out, int out_size, void* d_ws, size_t ws_size,
                              hipStream_t stream)
{
  (void)n_in; (void)out_size; (void)ws_size;
  const float* x0   = (const float*)d_in[0];
  const float* x1   = (const float*)d_in[1];
  const float* nz   = (const float*)d_in[2];
  const float* beta = (const float*)d_in[3];
  const int*  batch = (const int*)d_in[4];
  const int*  timev = (const int*)d_in[5];
  int T = in_sizes[3];
  int N = in_sizes[4];

  // workspace carve-up (floats/ints, all 4-byte aligned)
  float* sched = (float*)d_ws;                 // 4*T
  int*   start = (int*)(sched + 4 * T);        // GG+1
  float* Rm    = (float*)(start + (GG + 1));   // 9*GG
  float* cA    = Rm + 9 * GG;                  // 3*GG
  float* cB    = cA + 3 * GG;                  // 3*GG

  float* xt = (float*)d_out;                   // first output  [N,3]
  float* tg = xt + (size_t)3 * N;              // second output [N,3]

  const int blk = 256;
  int nq = N >> 2;

  i2sb_sched<<<1, 64, 0, stream>>>(beta, sched, T);
  i2sb_xt<<<(nq + blk - 1) / blk, blk, 0, stream>>>(
      (const float4*)x0, (const float4*)x1, (const float4*)nz,
      batch, timev, sched, (float4*)xt, N, T);
  i2sb_offs<<<(N + blk - 1) / blk, blk, 0, stream>>>(batch, start, N);
  i2sb_kabsch<<<GG, 256, 0, stream>>>(x0, xt, start, Rm, cA, cB);
  i2sb_target<<<(nq + blk - 1) / blk, blk, 0, stream>>>(
      (const float4*)x0, (const float4*)xt, batch, timev,
      sched + 3 * T, Rm, cA, cB, (float4*)tg, N);
}